// HuberCell_13657996001944
// MI455X (gfx1250) — compile-verified
//
#include <hip/hip_runtime.h>
#include <math.h>

// H = U = 64, system size N = 4096. Output = [Ltmp (4096) | Ptmp (4096)] fp32.

typedef float v2f __attribute__((ext_vector_type(2)));
typedef float v8f __attribute__((ext_vector_type(8)));

// ---- workspace layout (float offsets) ----
#define WS_STS    0        // S^T S           (64x64)
#define WS_R      4096     // CG residual
#define WS_P      8192     // CG direction
#define WS_X      12288    // CG solution (Xtmp flat)
#define WS_AP     16384    // A*p
#define WS_SP     20480    // (P @ StS) flat
#define WS_PART   24576    // per-WG partials (512)
#define WS_SCAL   25088    // rr scalar

#define NITER  192
#define NSWEEP 12

// ---------------- WMMA f32 16x16xK tile helper ----------------
// A 16x4 f32 frag: lanes 0-15 hold M=lane, K=0,1 in 2 VGPRs; lanes 16-31 K=2,3.
// B 4x16 f32 frag: VGPR0 = row K (=0 / 2 per lane-half), VGPR1 = next row, N=lane%16.
// C/D 16x16 f32:   VGPR r -> M = r + 8*(lane/16), N = lane%16.
__device__ __forceinline__ v8f wmma_tile(const float* __restrict__ Am, int lda, int transA,
                                         const float* __restrict__ Bm, int ldb, int transB,
                                         int I, int J, int K, v8f c)
{
  const int lane = threadIdx.x & 31;
  const int half = lane >> 4;
  const int l16  = lane & 15;
  const int m = I + l16;
  const int n = J + l16;
  for (int k0 = 0; k0 < K; k0 += 4) {
    const int ka = k0 + 2*half;
    v2f a, b;
    if (transA) { a[0] = Am[(ka+0)*lda + m]; a[1] = Am[(ka+1)*lda + m]; }
    else        { a[0] = Am[m*lda + ka+0];   a[1] = Am[m*lda + ka+1];   }
    if (transB) { b[0] = Bm[n*ldb + ka+0];   b[1] = Bm[n*ldb + ka+1];   }
    else        { b[0] = Bm[(ka+0)*ldb + n]; b[1] = Bm[(ka+1)*ldb + n]; }
    c = __builtin_amdgcn_wmma_f32_16x16x4_f32(false, a, false, b, (short)0, c,
                                              false, false);
  }
  return c;
}

__device__ __forceinline__ void store_tile(float* __restrict__ Dm, int ldd,
                                           int I, int J, v8f c)
{
  const int lane = threadIdx.x & 31;
  const int half = lane >> 4;
  const int l16  = lane & 15;
  for (int r = 0; r < 8; ++r)
    Dm[(I + r + 8*half)*ldd + (J + l16)] = c[r];
}

// -------- init: StS = S^T S (WMMA), rhs, rr0, and Sp0 = P0 @ StS (WMMA) --------
__global__ void __launch_bounds__(512) k_init(const float* __restrict__ S,
                       const float* __restrict__ x, const float* __restrict__ L,
                       const unsigned char* __restrict__ mask,
                       const float* __restrict__ thP, const float* __restrict__ rho_p,
                       float* __restrict__ ws)
{
  __shared__ float red[512];
  __shared__ float sts[4096];
  __shared__ float bs[4096];
  const int tid  = threadIdx.x;
  const int wave = tid >> 5;
  const int I = (wave >> 2)*16, J = (wave & 3)*16;
  {
    v8f c = {};
    c = wmma_tile(S, 64, 1, S, 64, 0, I, J, 64, c);   // S^T * S
    store_tile(sts, 64, I, J, c);
    store_tile(ws + WS_STS, 64, I, J, c);
  }
  const float rho = rho_p[0];
  float rr = 0.f;
  for (int k = 0; k < 8; ++k) {
    int i = tid + k*512;
    float b = rho*(L[i] - thP[i]) + (mask[i] ? x[i] : 0.f);
    ws[WS_R + i] = b; ws[WS_P + i] = b; ws[WS_X + i] = 0.f; bs[i] = b;
    rr += b*b;
  }
  red[tid] = rr; __syncthreads();
  for (int off = 256; off; off >>= 1) { if (tid < off) red[tid] += red[tid+off]; __syncthreads(); }
  if (tid == 0) ws[WS_SCAL] = red[0];
  __syncthreads();
  v8f c = {};
  c = wmma_tile(bs, 64, 0, sts, 64, 0, I, J, 64, c);  // Sp0 = P0 @ StS
  store_tile(ws + WS_SP, 64, I, J, c);
}

// ---- Ap = q.p + rho p + la2 Sp + la1 D p ; per-WG partial p.Ap ----
// p is staged LDS-direct via gfx1250 async-to-LDS (ASYNCcnt), no VGPR round-trip.
__global__ void __launch_bounds__(256) k_matvec(const float* __restrict__ D,
                         const unsigned char* __restrict__ mask,
                         const float* __restrict__ la1_p, const float* __restrict__ la2_p,
                         const float* __restrict__ rho_p, float* __restrict__ ws)
{
  __shared__ float ps[4096];
  __shared__ float wred[8];
  const int tid = threadIdx.x;
  {
    const unsigned ldsbase = (unsigned)(size_t)(void*)ps;     // low 32 bits = LDS offset
    const unsigned long long pa = (unsigned long long)(size_t)(ws + WS_P);
#pragma unroll
    for (int k = 0; k < 4; ++k) {
      unsigned off = (unsigned)((tid + k*256) * 16);
      asm volatile("global_load_async_to_lds_b128 %0, %1, %2"
                   :: "v"(ldsbase + off), "v"(off), "s"(pa) : "memory");
    }
    asm volatile("s_wait_asynccnt 0" ::: "memory");
  }
  __syncthreads();
  const int lane = tid & 31, wave = tid >> 5;
  const int row  = blockIdx.x*8 + wave;
  const float* Drow = D + (size_t)row * 4096;
  float acc = 0.f;
  for (int j = lane*4; j < 4096; j += 128) {
    __builtin_prefetch(Drow + j + 512, 0, 1);
    const float4 d = *(const float4*)(Drow + j);
    acc += d.x*ps[j] + d.y*ps[j+1] + d.z*ps[j+2] + d.w*ps[j+3];
  }
  for (int off = 16; off; off >>= 1) acc += __shfl_down(acc, off);
  if (lane == 0) {
    float pr  = ps[row];
    float apv = la1_p[0]*acc + rho_p[0]*pr + (mask[row] ? pr : 0.f)
              + la2_p[0]*ws[WS_SP + row];
    ws[WS_AP + row] = apv;
    wred[wave] = pr * apv;
  }
  __syncthreads();
  if (tid == 0) {
    float s = 0.f;
    for (int w = 0; w < 8; ++w) s += wred[w];
    ws[WS_PART + blockIdx.x] = s;
  }
}

// ---- fused CG update + next-iteration Sp = P @ StS (WMMA from LDS p) ----
__global__ void __launch_bounds__(512) k_update_sp(float* __restrict__ ws)
{
  __shared__ float red[512];
  __shared__ float ps[4096];
  const int tid = threadIdx.x;
  red[tid] = ws[WS_PART + tid];
  __syncthreads();
  for (int off = 256; off; off >>= 1) { if (tid < off) red[tid] += red[tid+off]; __syncthreads(); }
  const float pAp    = red[0];
  const float rr_old = ws[WS_SCAL];
  __syncthreads();
  const float alpha = rr_old / (pAp + 1e-30f);
  float rn = 0.f;
  for (int k = 0; k < 8; ++k) {
    int i = tid + k*512;
    ws[WS_X + i] += alpha * ws[WS_P + i];
    float rv = ws[WS_R + i] - alpha * ws[WS_AP + i];
    ws[WS_R + i] = rv;
    rn += rv*rv;
  }
  red[tid] = rn; __syncthreads();
  for (int off = 256; off; off >>= 1) { if (tid < off) red[tid] += red[tid+off]; __syncthreads(); }
  const float rr_new = red[0];
  const float beta   = rr_new / (rr_old + 1e-30f);
  for (int k = 0; k < 8; ++k) {
    int i = tid + k*512;
    float pn = ws[WS_R + i] + beta * ws[WS_P + i];
    ws[WS_P + i] = pn;
    ps[i] = pn;
  }
  if (tid == 0) ws[WS_SCAL] = rr_new;
  __syncthreads();
  const int wave = tid >> 5;
  const int I = (wave >> 2)*16, J = (wave & 3)*16;
  v8f c = {};
  c = wmma_tile(ps, 64, 0, ws + WS_STS, 64, 0, I, J, 64, c);
  store_tile(ws + WS_SP, 64, I, J, c);
}

// ---- fused SVT: B = X+thP; G = B^T B (WMMA); parallel Jacobi eig;
//      Ltmp = (B V) diag(relu(s-tau)/s) V^T (2 WMMA GEMMs); Ptmp epilogue.
//      All 64x64 operands live in LDS; 32 waves (waves>=16 duplicate WMMA tiles,
//      only waves<16 store, keeping EXEC uniform around WMMA).
__global__ void __launch_bounds__(1024) k_post(const float* __restrict__ thP,
                                               const float* __restrict__ v_p,
                                               const float* __restrict__ gamma_p,
                                               const float* __restrict__ neta_p,
                                               float* __restrict__ out,
                                               float* __restrict__ ws)
{
  __shared__ float Bs[4096];
  __shared__ float Gs[4096];   // reused as C1 = B@V after shrink factors extracted
  __shared__ float Vs[4096];
  __shared__ float fs[64];
  __shared__ int top[32], bot[32];
  const int tid  = threadIdx.x;
  const int lane = tid & 31, wave = tid >> 5;
  for (int k = 0; k < 4; ++k) {
    int i = tid + k*1024;
    Bs[i] = ws[WS_X + i] + thP[i];
    Vs[i] = ((i >> 6) == (i & 63)) ? 1.f : 0.f;
  }
  if (tid < 32) { top[tid] = 2*tid; bot[tid] = 2*tid + 1; }
  __syncthreads();
  const int w16 = wave & 15;
  const int I = (w16 >> 2)*16, J = (w16 & 3)*16;
  {
    v8f c = {};
    c = wmma_tile(Bs, 64, 1, Bs, 64, 0, I, J, 64, c);      // G = B^T B
    if (wave < 16) store_tile(Gs, 64, I, J, c);
  }
  __syncthreads();
  // parallel two-sided Jacobi: 32 disjoint pairs per round, round-robin tournament
  for (int sweep = 0; sweep < NSWEEP; ++sweep) {
    for (int rnd = 0; rnd < 63; ++rnd) {
      int pp = top[wave], qq = bot[wave];
      if (pp > qq) { int t = pp; pp = qq; qq = t; }
      const float app = Gs[pp*64+pp], aqq = Gs[qq*64+qq], apq = Gs[pp*64+qq];
      float c = 1.f, s = 0.f;
      if (fabsf(apq) > 1e-20f) {
        float theta = (aqq - app) / (2.f * apq);
        float t = 1.f / (fabsf(theta) + sqrtf(1.f + theta*theta));
        t = (theta < 0.f) ? -t : t;
        c = rsqrtf(1.f + t*t);
        s = t * c;
      }
      // column updates: columns pp,qq disjoint across pairs -> no barrier needed
      for (int rep = 0; rep < 2; ++rep) {
        int i = lane + 32*rep;
        float gp = Gs[i*64+pp], gq = Gs[i*64+qq];
        Gs[i*64+pp] = c*gp - s*gq; Gs[i*64+qq] = s*gp + c*gq;
        float vp = Vs[i*64+pp], vq = Vs[i*64+qq];
        Vs[i*64+pp] = c*vp - s*vq; Vs[i*64+qq] = s*vp + c*vq;
      }
      __syncthreads();           // all column updates before row updates
      for (int rep = 0; rep < 2; ++rep) {
        int i = lane + 32*rep;
        float gp = Gs[pp*64+i], gq = Gs[qq*64+i];
        Gs[pp*64+i] = c*gp - s*gq; Gs[qq*64+i] = s*gp + c*gq;
      }
      __syncthreads();
      if (tid == 0) {            // round-robin rotation (top[0] fixed)
        int saved = top[31];
        for (int i2 = 31; i2 >= 2; --i2) top[i2] = top[i2-1];
        top[1] = bot[0];
        for (int i2 = 0; i2 < 31; ++i2) bot[i2] = bot[i2+1];
        bot[31] = saved;
      }
      __syncthreads();
    }
  }
  if (tid < 64) {                // shrink factors relu(s - tau)/s from eigvals
    float tau = gamma_p[0] / (1.f + expf(-v_p[0]));   // sigmoid(v)*coef_gamma
    float d   = Gs[tid*64 + tid];
    float sv  = sqrtf(fmaxf(d, 0.f));
    fs[tid]   = (sv > tau) ? (sv - tau) / fmaxf(sv, 1e-20f) : 0.f;
  }
  __syncthreads();               // fs done & Gs diag read before Gs is overwritten
  {
    v8f c1 = {};
    c1 = wmma_tile(Bs, 64, 0, Vs, 64, 0, I, J, 64, c1);    // C1 = B @ V
    if (wave < 16) store_tile(Gs, 64, I, J, c1);           // Gs now holds C1
  }
  __syncthreads();
  for (int k = 0; k < 4; ++k) {
    int i = tid + k*1024;
    Gs[i] *= fs[i & 63];                                   // scale columns
  }
  __syncthreads();
  {
    v8f c2 = {};
    c2 = wmma_tile(Gs, 64, 0, Vs, 64, 1, I, J, 64, c2);    // Ltmp = C1 @ V^T
    if (wave < 16) {
      const int half = lane >> 4, l16 = lane & 15;
      const float neta = neta_p[0];
      for (int r = 0; r < 8; ++r) {
        int row = I + r + 8*half, col = J + l16;
        int idx = row*64 + col;
        float lt = c2[r];
        out[idx]        = lt;                                        // Ltmp
        out[4096 + idx] = thP[idx] + neta * (ws[WS_X + idx] - lt);   // Ptmp
      }
    }
  }
}

extern "C" void kernel_launch(void* const* d_in, const int* in_sizes, int n_in,
                              void* d_out, int out_size, void* d_ws, size_t ws_size,
                              hipStream_t stream)
{
  (void)in_sizes; (void)n_in; (void)out_size; (void)ws_size;
  const float*         x    = (const float*)d_in[0];
  const float*         L    = (const float*)d_in[1];
  const unsigned char* mask = (const unsigned char*)d_in[2];
  const float*         D    = (const float*)d_in[3];
  const float*         thP  = (const float*)d_in[4];
  const float*         v    = (const float*)d_in[5];
  const float*         cg   = (const float*)d_in[6];
  const float*         neta = (const float*)d_in[7];
  const float*         la1  = (const float*)d_in[8];
  const float*         la2  = (const float*)d_in[9];
  const float*         rho  = (const float*)d_in[10];
  const float*         S    = (const float*)d_in[11];
  float* out = (float*)d_out;
  float* ws  = (float*)d_ws;

  k_init<<<1, 512, 0, stream>>>(S, x, L, mask, thP, rho, ws);
  for (int it = 0; it < NITER; ++it) {
    k_matvec   <<<512, 256, 0, stream>>>(D, mask, la1, la2, rho, ws);
    k_update_sp<<<1,   512, 0, stream>>>(ws);
  }
  k_post<<<1, 1024, 0, stream>>>(thP, v, cg, neta, out, ws);
}